// MultiHeadedAttention_88682484728307
// MI455X (gfx1250) — compile-verified
//
#include <hip/hip_runtime.h>
#include <hip/hip_bf16.h>

typedef __attribute__((ext_vector_type(2))) float v2f;
typedef __attribute__((ext_vector_type(8))) float v8f;

#define H_HEADS 16
#define D_HEAD  64
#define E_DIM   1024
#define S_LEN   2048
#define M_DIM   4096        // B*S
#define K_DIM   1024

__device__ inline v8f wmma4(v2f a, v2f b, v8f c) {
    // D = A(16x4 f32) * B(4x16 f32) + C(16x16 f32)
    return __builtin_amdgcn_wmma_f32_16x16x4_f32(
        false, a, false, b, (short)0, c, false, false);
}

__device__ inline v8f zero8() {
    v8f z;
#pragma unroll
    for (int i = 0; i < 8; ++i) z[i] = 0.0f;
    return z;
}

// CDNA5 async global->LDS copy (ASYNCcnt-tracked, no VGPR staging).
__device__ inline void async_ld_b128(unsigned lds_addr, const float* gaddr) {
    asm volatile("global_load_async_to_lds_b128 %0, %1, off"
                 :: "v"(lds_addr), "v"(gaddr) : "memory");
}
__device__ inline void wait_async0() {
#if __has_builtin(__builtin_amdgcn_s_wait_asynccnt)
    __builtin_amdgcn_s_wait_asynccnt(0);
#else
    asm volatile("s_wait_asynccnt 0x0" ::: "memory");
#endif
}
__device__ inline unsigned lds_off(const void* p) {
    return (unsigned)(uintptr_t)p;   // low 32 bits of generic LDS ptr = DS offset
}

// ---------------------------------------------------------------------------
// GEMM: out = X[M,1024] @ W[1024,1024]^T + bias (+ residual). headLayout=1
// writes [B,H,S,D] (n -> h*64+d), headLayout=0 writes row-major [M,N].
// Block: 256 thr (8 waves). Tile: 128(M) x 64(N). K double-buffered 32-wide
// in LDS via async global->LDS loads; frag loads software-pipelined.
// ---------------------------------------------------------------------------
#define TM 128
#define TN 64
#define TK 32
#define NCHUNK (K_DIM / TK)   // 32
#define XS_STRIDE 36          // 144B rows: 16B aligned, conflict-free frags
#define WS_STRIDE 36
#define XBUF_BYTES (TM * XS_STRIDE * 4)
#define WBUF_BYTES (TN * WS_STRIDE * 4)

__global__ __launch_bounds__(256, 1) void gemm_xwT(
    const float* __restrict__ X, const float* __restrict__ W,
    const float* __restrict__ bias, const float* __restrict__ resid,
    float* __restrict__ out, int headLayout)
{
    __shared__ float Xs[2][TM * XS_STRIDE];
    __shared__ float Ws[2][TN * WS_STRIDE];

    const int tid  = threadIdx.x;
    const int wave = tid >> 5;
    const int lane = tid & 31;
    const int h    = lane >> 4;     // wave half
    const int q    = lane & 15;     // m (or n) within tile

    const int m0 = (blockIdx.x & 31) * TM;    // 32 m-tiles
    const int n0 = (blockIdx.x >> 5) * TN;    // 16 n-tiles

    // per-thread staging: walking global pointers + LDS byte offsets
    const float* px[4];
    const float* pw[2];
    unsigned xoff[4], woff[2];
    const unsigned xbase = lds_off(&Xs[0][0]);
    const unsigned wbase = lds_off(&Ws[0][0]);
#pragma unroll
    for (int j = 0; j < 4; ++j) {
        int idx = tid + j * 256;
        int r = idx >> 3, c = (idx & 7) << 2;
        px[j]   = X + (size_t)(m0 + r) * K_DIM + c;
        xoff[j] = (unsigned)(r * XS_STRIDE + c) * 4u;
    }
#pragma unroll
    for (int j = 0; j < 2; ++j) {
        int idx = tid + j * 256;
        int r = idx >> 3, c = (idx & 7) << 2;
        pw[j]   = W + (size_t)(n0 + r) * K_DIM + c;
        woff[j] = (unsigned)(r * WS_STRIDE + c) * 4u;
    }

    // stage chunk 0 into buffer 0
#pragma unroll
    for (int j = 0; j < 4; ++j) { async_ld_b128(xbase + xoff[j], px[j]); px[j] += TK; }
#pragma unroll
    for (int j = 0; j < 2; ++j) { async_ld_b128(wbase + woff[j], pw[j]); pw[j] += TK; }
    wait_async0();
    __syncthreads();

    v8f acc[4];
#pragma unroll
    for (int s = 0; s < 4; ++s) acc[s] = zero8();

    for (int chunk = 0; chunk < NCHUNK; ++chunk) {
        const int buf = chunk & 1;
        // async-stage next chunk into the other buffer (overlaps compute);
        // safe: previous trailing barrier means no wave still reads buf^1
        if (chunk + 1 < NCHUNK) {
            const unsigned nbx = xbase + (buf ^ 1) * XBUF_BYTES;
            const unsigned nbw = wbase + (buf ^ 1) * WBUF_BYTES;
#pragma unroll
            for (int j = 0; j < 4; ++j) { async_ld_b128(nbx + xoff[j], px[j]); px[j] += TK; }
#pragma unroll
            for (int j = 0; j < 2; ++j) { async_ld_b128(nbw + woff[j], pw[j]); pw[j] += TK; }
        }

        // compute: software-pipelined fragment loads
        const float* xb = &Xs[buf][(wave * 16 + q) * XS_STRIDE + 2 * h];
        const float* wb = &Ws[buf][q * WS_STRIDE + 2 * h];
        v2f a_cur = *(const v2f*)xb;
        v2f b_cur[4];
#pragma unroll
        for (int s = 0; s < 4; ++s)
            b_cur[s] = *(const v2f*)(wb + s * 16 * WS_STRIDE);
#pragma unroll
        for (int t = 0; t < TK / 4; ++t) {
            v2f a_nxt{}; v2f b_nxt[4];
            if (t + 1 < TK / 4) {
                a_nxt = *(const v2f*)(xb + 4 * (t + 1));
#pragma unroll
                for (int s = 0; s < 4; ++s)
                    b_nxt[s] = *(const v2f*)(wb + s * 16 * WS_STRIDE + 4 * (t + 1));
            }
#pragma unroll
            for (int s = 0; s < 4; ++s)
                acc[s] = wmma4(a_cur, b_cur[s], acc[s]);
            if (t + 1 < TK / 4) {
                a_cur = a_nxt;
#pragma unroll
                for (int s = 0; s < 4; ++s) b_cur[s] = b_nxt[s];
            }
        }

        if (chunk + 1 < NCHUNK) {
            wait_async0();     // my async loads landed in LDS
            __syncthreads();   // everyone's landed -> next compute may read
        }
    }

    // epilogue: C frag c[r] = C[r + 8h][n], n = q
#pragma unroll
    for (int s = 0; s < 4; ++s) {
        int n = n0 + s * 16 + q;
        float bn = bias[n];
#pragma unroll
        for (int r = 0; r < 8; ++r) {
            int m = m0 + wave * 16 + r + 8 * h;
            float v = acc[s][r] + bn;
            if (headLayout) {
                int bb = m >> 11, ss = m & (S_LEN - 1);
                int hh = n >> 6,  dd = n & 63;
                out[(size_t)((bb * H_HEADS + hh) * S_LEN + ss) * D_HEAD + dd] = v;
            } else {
                if (resid) v += resid[(size_t)m * E_DIM + n];
                out[(size_t)m * E_DIM + n] = v;
            }
        }
    }
}

// ---------------------------------------------------------------------------
// Flash attention: Q,K,V in [B*H, S, 64]. Block = 8 waves; each wave owns a
// 16-row Q tile; K/V 16x64 tiles double-buffered in LDS via async loads.
// ---------------------------------------------------------------------------
#define KV_STRIDE 68   // 272B rows: 16B aligned, conflict-free score reads
#define PS_STRIDE 18
#define NKT (S_LEN / 16)        // 128 key tiles
#define KVBUF_BYTES (16 * KV_STRIDE * 4)

__global__ __launch_bounds__(256, 1) void attn_kernel(
    const float* __restrict__ Q, const float* __restrict__ Kg,
    const float* __restrict__ Vg, const int* __restrict__ mask,
    float* __restrict__ ctx)
{
    __shared__ float Ks[2][16 * KV_STRIDE];
    __shared__ float Vs[2][16 * KV_STRIDE];
    __shared__ float Ps[8 * 16 * PS_STRIDE];

    const int tid  = threadIdx.x;
    const int wave = tid >> 5;
    const int lane = tid & 31;
    const int h    = lane >> 4;
    const int n    = lane & 15;

    const int bh = blockIdx.x;                 // b*H + head
    const int q0 = blockIdx.y * 128 + wave * 16;

    const float* Qb = Q + (size_t)bh * S_LEN * D_HEAD;

    // preload Q fragments: a[t][j] = Q[q0+m][4t + 2h + j], m = lane%16
    v2f qf[16];
#pragma unroll
    for (int t = 0; t < 16; ++t)
        qf[t] = *(const v2f*)(Qb + (size_t)(q0 + n) * D_HEAD + 4 * t + 2 * h);

    float mi[8], li[8];
    v8f o[4];
#pragma unroll
    for (int r = 0; r < 8; ++r) { mi[r] = -3.0e38f; li[r] = 0.0f; }
#pragma unroll
    for (int s = 0; s < 4; ++s) o[s] = zero8();

    // K/V staging: 16 rows x 16 float4-cols, one async b128 each per tile
    const int srow = tid >> 4;
    const int sc4  = (tid & 15) << 2;
    const float* pk = Kg + (size_t)bh * S_LEN * D_HEAD + (size_t)srow * D_HEAD + sc4;
    const float* pv = Vg + (size_t)bh * S_LEN * D_HEAD + (size_t)srow * D_HEAD + sc4;
    const unsigned kvoff = (unsigned)(srow * KV_STRIDE + sc4) * 4u;
    const unsigned kbase = lds_off(&Ks[0][0]) + kvoff;
    const unsigned vbase = lds_off(&Vs[0][0]) + kvoff;

    async_ld_b128(kbase, pk); pk += 16 * D_HEAD;
    async_ld_b128(vbase, pv); pv += 16 * D_HEAD;
    wait_async0();
    __syncthreads();

    for (int kt = 0; kt < NKT; ++kt) {
        const int buf = kt & 1;
        const int j0  = kt * 16;
        if (kt + 1 < NKT) {   // async-stage next K/V tile into other buffer
            async_ld_b128(kbase + (buf ^ 1) * KVBUF_BYTES, pk); pk += 16 * D_HEAD;
            async_ld_b128(vbase + (buf ^ 1) * KVBUF_BYTES, pv); pv += 16 * D_HEAD;
        }

        // scores = Q @ K^T : B-frag b[j] = K[j0+n][4t+2h+j], pipelined
        v8f c = zero8();
        {
            const float* kb = &Ks[buf][n * KV_STRIDE + 2 * h];
            v2f bc = *(const v2f*)kb;
#pragma unroll
            for (int t = 0; t < 16; ++t) {
                v2f bn_{};
                if (t + 1 < 16) bn_ = *(const v2f*)(kb + 4 * (t + 1));
                c = wmma4(qf[t], bc, c);
                bc = bn_;
            }
        }

        // mask + online softmax; c[r] = score[r+8h][n]
#pragma unroll
        for (int r = 0; r < 8; ++r) {
            float s = c[r] * 0.125f;                 // 1/sqrt(64)
            int mrow = q0 + r + 8 * h;
            int mv = mask[((size_t)mrow << 11) + j0 + n];
            s = (mv == 0) ? -1.0e9f : s;
            float rm = s;
            rm = fmaxf(rm, __shfl_xor(rm, 1, 16));
            rm = fmaxf(rm, __shfl_xor(rm, 2, 16));
            rm = fmaxf(rm, __shfl_xor(rm, 4, 16));
            rm = fmaxf(rm, __shfl_xor(rm, 8, 16));
            float nm   = fmaxf(mi[r], rm);
            float corr = __expf(mi[r] - nm);
            float pe   = __expf(s - nm);
            float rs = pe;
            rs += __shfl_xor(rs, 1, 16);
            rs += __shfl_xor(rs, 2, 16);
            rs += __shfl_xor(rs, 4, 16);
            rs += __shfl_xor(rs, 8, 16);
            li[r] = li[r] * corr + rs;
            mi[r] = nm;
#pragma unroll
            for (int s4 = 0; s4 < 4; ++s4) o[s4][r] *= corr;
            Ps[(wave * 16 + r + 8 * h) * PS_STRIDE + n] = pe;
        }
        __syncthreads();   // Ps visible to whole wave group

        // ctx += P @ V : A-frag from Ps, B-frag from Vs[buf] rows
#pragma unroll
        for (int t = 0; t < 4; ++t) {
            int kk = 4 * t + 2 * h;
            v2f a = *(const v2f*)&Ps[(wave * 16 + n) * PS_STRIDE + kk];
#pragma unroll
            for (int s4 = 0; s4 < 4; ++s4) {
                v2f b;
                b.x = Vs[buf][kk * KV_STRIDE + s4 * 16 + n];
                b.y = Vs[buf][(kk + 1) * KV_STRIDE + s4 * 16 + n];
                o[s4] = wmma4(a, b, o[s4]);
            }
        }

        if (kt + 1 < NKT) wait_async0();   // next tile landed in LDS
        __syncthreads();   // all waves done: Ps reusable, buf^1 readable
    }

    // finalize + store ctx into [B, S, E] (col = head*64 + d)
    const int bb = bh / H_HEADS;
    const int hd = bh % H_HEADS;
#pragma unroll
    for (int r = 0; r < 8; ++r) {
        float inv = 1.0f / li[r];
#pragma unroll
        for (int s4 = 0; s4 < 4; ++s4) {
            float v = o[s4][r] * inv;
            int m = q0 + r + 8 * h;
            ctx[(size_t)(bb * S_LEN + m) * E_DIM + hd * D_HEAD + s4 * 16 + n] = v;
        }
    }
}

// ---------------------------------------------------------------------------
// Row-wise LayerNorm over E=1024. One block per row.
// ---------------------------------------------------------------------------
__global__ __launch_bounds__(256) void ln_kernel(
    const float* __restrict__ y, const float* __restrict__ gamma,
    const float* __restrict__ beta, float* __restrict__ out)
{
    const int row = blockIdx.x;
    const int tid = threadIdx.x;
    const float* yr = y + (size_t)row * E_DIM;
    float4 v = *(const float4*)(yr + tid * 4);
    float s  = v.x + v.y + v.z + v.w;
    float ss = v.x * v.x + v.y * v.y + v.z * v.z + v.w * v.w;
#pragma unroll
    for (int off = 1; off < 32; off <<= 1) {
        s  += __shfl_xor(s,  off, 32);
        ss += __shfl_xor(ss, off, 32);
    }
    __shared__ float rs[8], rss[8];
    if ((tid & 31) == 0) { rs[tid >> 5] = s; rss[tid >> 5] = ss; }
    __syncthreads();
    float S1 = 0.0f, S2 = 0.0f;
#pragma unroll
    for (int i = 0; i < 8; ++i) { S1 += rs[i]; S2 += rss[i]; }
    float mu  = S1 * (1.0f / E_DIM);
    float var = S2 * (1.0f / E_DIM) - mu * mu;
    float inv = rsqrtf(var + 1e-5f);
    float4 g  = *(const float4*)(gamma + tid * 4);
    float4 bt = *(const float4*)(beta + tid * 4);
    float4 o;
    o.x = (v.x - mu) * inv * g.x + bt.x;
    o.y = (v.y - mu) * inv * g.y + bt.y;
    o.z = (v.z - mu) * inv * g.z + bt.z;
    o.w = (v.w - mu) * inv * g.w + bt.w;
    *(float4*)(out + (size_t)row * E_DIM + tid * 4) = o;
}

// ---------------------------------------------------------------------------
extern "C" void kernel_launch(void* const* d_in, const int* in_sizes, int n_in,
                              void* d_out, int out_size, void* d_ws, size_t ws_size,
                              hipStream_t stream) {
    const float* x     = (const float*)d_in[0];
    const int*   mask  = (const int*)d_in[1];
    const float* Wq    = (const float*)d_in[2];
    const float* bq    = (const float*)d_in[3];
    const float* Wk    = (const float*)d_in[4];
    const float* bk    = (const float*)d_in[5];
    const float* Wv    = (const float*)d_in[6];
    const float* bv    = (const float*)d_in[7];
    const float* Wo    = (const float*)d_in[8];
    const float* bo    = (const float*)d_in[9];
    const float* gamma = (const float*)d_in[10];
    const float* beta  = (const float*)d_in[11];

    const size_t plane = (size_t)M_DIM * E_DIM;   // 4M floats

    float* qws = (float*)d_ws;
    float* kws = qws + plane;
    float* vws = kws + plane;
    float* cws = vws + plane;
    float* yws = cws + plane;

    dim3 blk(256);
    dim3 gG((M_DIM / TM) * (E_DIM / TN));   // 32 * 16 = 512 blocks

    gemm_xwT<<<gG, blk, 0, stream>>>(x, Wq, bq, nullptr, qws, 1);
    gemm_xwT<<<gG, blk, 0, stream>>>(x, Wk, bk, nullptr, kws, 1);
    gemm_xwT<<<gG, blk, 0, stream>>>(x, Wv, bv, nullptr, vws, 1);

    dim3 gA(2 * H_HEADS, S_LEN / 128);      // (32, 16)
    attn_kernel<<<gA, blk, 0, stream>>>(qws, kws, vws, mask, cws);

    gemm_xwT<<<gG, blk, 0, stream>>>(cws, Wo, bo, x, yws, 0);

    ln_kernel<<<M_DIM, blk, 0, stream>>>(yws, gamma, beta, (float*)d_out);
}